// MemoryEfficientVoxelizer_57062935495125
// MI455X (gfx1250) — compile-verified
//
#include <hip/hip_runtime.h>
#include <hip/hip_bf16.h>

// MemoryEfficientVoxelizer for MI455X (gfx1250, wave32).
// density = sum_n contrib[n,v]; feats = (contrib^T @ features) / density.
// The einsum is mapped onto V_WMMA_F32_16X16X4_F32: M=16 voxels, N=16 features
// (x2 tiles for F=32), K=4 gaussians per instruction, K-loop of 128 steps.

typedef __attribute__((ext_vector_type(2))) float v2f;
typedef __attribute__((ext_vector_type(8))) float v8f;

#define N_G     512
#define FDIM    32
#define NVOX    76800      // 80*80*12
#define GSTRIDE 12         // floats per gaussian record (3 x float4)

__global__ __launch_bounds__(128) void voxelizer_wmma_kernel(
    const float* __restrict__ means3d,     // (512,3)
    const float* __restrict__ opacities,   // (512,1)
    const float* __restrict__ cov,         // (512,3,3)
    const float* __restrict__ feats,       // (512,32)
    float* __restrict__ out)               // [density 76800 | feats 76800*32]
{
    __shared__ __align__(16) float sG[N_G * GSTRIDE]; // 24 KB gaussian records
    __shared__ __align__(16) float sF[N_G * FDIM];    // 64 KB features

    const int tid = threadIdx.x;

    // ---- stage features into LDS (4096 float4 / 128 threads) ----
    {
        const float4* src = (const float4*)feats;
        float4*       dst = (float4*)sF;
        #pragma unroll
        for (int i = tid; i < (N_G * FDIM) / 4; i += 128) dst[i] = src[i];
    }

    // ---- per-gaussian precompute: Sigma^-1 (symmetric adjugate), keep mask, r^2 ----
    for (int g = tid; g < N_G; g += 128) {
        float mx = means3d[g * 3 + 0];
        float my = means3d[g * 3 + 1];
        float mz = means3d[g * 3 + 2];
        float op = opacities[g];
        const float* c = cov + g * 9;
        float c00 = c[0], c01 = c[1], c02 = c[2];
        float c11 = c[4], c12 = c[5], c22 = c[8];
        float s0 = sqrtf(c00), s1 = sqrtf(c11), s2 = sqrtf(c22);
        bool keep = (mx + 3.f * s0 > -20.f) && (my + 3.f * s1 > -20.f) &&
                    (mz + 3.f * s2 >  -2.f) && (mx - 3.f * s0 <  20.f) &&
                    (my - 3.f * s1 <  20.f) && (mz - 3.f * s2 <  4.4f) &&
                    (op > 1e-4f);
        float opEff = keep ? op : 0.f;
        float sm = fmaxf(s0, fmaxf(s1, s2));
        float r2 = 9.f * sm * sm;                  // (3*sigma_max)^2
        float m00 = c11 * c22 - c12 * c12;
        float m01 = c02 * c12 - c01 * c22;
        float m02 = c01 * c12 - c02 * c11;
        float det = c00 * m00 + c01 * m01 + c02 * m02;
        float inv = 1.f / det;
        float i00 = m00 * inv, i01 = m01 * inv, i02 = m02 * inv;
        float i11 = (c00 * c22 - c02 * c02) * inv;
        float i12 = (c01 * c02 - c00 * c12) * inv;
        float i22 = (c00 * c11 - c01 * c01) * inv;
        float* p = sG + g * GSTRIDE;
        p[0] = mx;  p[1] = my;  p[2] = mz;  p[3]  = opEff;
        p[4] = i00; p[5] = i01; p[6] = i02; p[7]  = r2;
        p[8] = i11; p[9] = i12; p[10] = i22; p[11] = 0.f;
    }
    __syncthreads();

    // ---- per-wave 16-voxel x 32-feature tile ----
    const int lane    = tid & 31;
    const int wave    = tid >> 5;
    const int tile    = blockIdx.x * 4 + wave;   // 1200*4 = 4800 tiles, exact
    const int voxBase = tile * 16;
    const int m       = lane & 15;               // voxel row (M) / feature col (N)
    const int half    = lane >> 4;               // wave32 half selects K pair

    // lane's voxel position (row M = m of this tile)
    int v  = voxBase + m;
    int iz = v % 12;
    int t2 = v / 12;
    int iy = t2 % 80;
    int ix = t2 / 80;
    float px = ((float)ix + 0.5f) * 0.5f - 20.f;
    float py = ((float)iy + 0.5f) * 0.5f - 20.f;
    float pz = ((float)iz + 0.5f) * 0.5f -  2.f;

    v8f acc0 = {};   // features 0..15
    v8f acc1 = {};   // features 16..31
    float dens = 0.f;

    const int gOff = half * 2;
    #pragma unroll 4
    for (int k = 0; k < N_G; k += 4) {
        int g0 = k + gOff;
        int g1 = g0 + 1;
        const float4* p0 = (const float4*)(sG + g0 * GSTRIDE);
        float4 A0 = p0[0], B0 = p0[1], C0 = p0[2];
        const float4* p1 = (const float4*)(sG + g1 * GSTRIDE);
        float4 A1 = p1[0], B1 = p1[1], C1 = p1[2];

        // contrib(g0, voxel m)
        float dx = A0.x - px, dy = A0.y - py, dz = A0.z - pz;
        float maha = B0.x * dx * dx + C0.x * dy * dy + C0.z * dz * dz +
                     2.f * (B0.y * dx * dy + B0.z * dx * dz + C0.y * dy * dz);
        float d2 = dx * dx + dy * dy + dz * dz;
        float e  = A0.w * __expf(-0.5f * maha);
        float a0 = (d2 < B0.w) ? e : 0.f;

        // contrib(g1, voxel m)
        dx = A1.x - px; dy = A1.y - py; dz = A1.z - pz;
        maha = B1.x * dx * dx + C1.x * dy * dy + C1.z * dz * dz +
               2.f * (B1.y * dx * dy + B1.z * dx * dz + C1.y * dy * dz);
        d2 = dx * dx + dy * dy + dz * dz;
        e  = A1.w * __expf(-0.5f * maha);
        float a1 = (d2 < B1.w) ? e : 0.f;

        dens += a0 + a1;

        // A-matrix 16x4 f32 layout: lanes 0-15 carry K=0,1; lanes 16-31 K=2,3.
        v2f Amat; Amat[0] = a0; Amat[1] = a1;

        // B-matrix 4x16: N striped over lane groups, K pair per VGPR (mirrors A).
        v2f Bm0, Bm1;
        Bm0[0] = sF[g0 * FDIM + m];
        Bm0[1] = sF[g1 * FDIM + m];
        Bm1[0] = sF[g0 * FDIM + 16 + m];
        Bm1[1] = sF[g1 * FDIM + 16 + m];

        acc0 = __builtin_amdgcn_wmma_f32_16x16x4_f32(
            false, Amat, false, Bm0, (short)0, acc0, false, false);
        acc1 = __builtin_amdgcn_wmma_f32_16x16x4_f32(
            false, Amat, false, Bm1, (short)0, acc1, false, false);
    }

    // ---- density: combine the two wave32 halves (K split) ----
    float densFull = dens + __shfl_xor(dens, 16, 32);
    if (lane < 16) out[voxBase + lane] = densFull;

    // ---- normalize + store features; C/D layout: VGPR i -> M = i + 8*half ----
    float* outF = out + NVOX;
    #pragma unroll
    for (int i = 0; i < 8; ++i) {
        int M = i + half * 8;
        float dm = __shfl(densFull, M, 32);
        float s  = 1.f / fmaxf(dm, 1e-6f);
        int vox = voxBase + M;
        outF[vox * FDIM + m]      = acc0[i] * s;
        outF[vox * FDIM + 16 + m] = acc1[i] * s;
    }
}

extern "C" void kernel_launch(void* const* d_in, const int* in_sizes, int n_in,
                              void* d_out, int out_size, void* d_ws, size_t ws_size,
                              hipStream_t stream) {
    const float* means3d   = (const float*)d_in[0];
    const float* opacities = (const float*)d_in[1];
    const float* cov       = (const float*)d_in[2];
    const float* feats     = (const float*)d_in[3];
    float* out = (float*)d_out;
    // 1200 blocks x 128 threads (4 waves) x 16 voxels/wave = 76800 voxels exactly.
    voxelizer_wmma_kernel<<<1200, 128, 0, stream>>>(means3d, opacities, cov, feats, out);
}